// gen_En_5815385718889
// MI455X (gfx1250) — compile-verified
//
#include <hip/hip_runtime.h>
#include <cstdint>
#include <cstddef>

// Problem constants (from reference)
constexpr int NGRID   = 32;          // N
constexpr int MODES   = 4;
constexpr int EY_SZ   = 192;         // EY_SIZE
constexpr int EY_PL   = EY_SZ * EY_SZ;   // 36864 floats per (k,m) plane
constexpr int OUT_RES = 32;
constexpr int TOTAL   = 1184;        // (N + 2*KNN + 1) * OUT_RES
constexpr int NTILE   = TOTAL / 32;  // 37 tiles per dim

// Stage: 4 modes x 32 x 32 floats = 16 KB, triple buffered in LDS (48 KB).
constexpr int STAGE_FLOATS = MODES * 32 * 32;     // 4096
constexpr int STAGE_BYTES  = STAGE_FLOATS * 4;    // 16384
constexpr int NBUF         = 3;

__global__ __launch_bounds__(256) void gen_en_tile_kernel(
    const float* __restrict__ U,
    const float* __restrict__ neff,
    const float* __restrict__ Ey,
    float* __restrict__ En)
{
    __shared__ __align__(16) float smem[NBUF * STAGE_FLOATS];   // 48 KB of 320 KB

    const int t  = threadIdx.x;          // 0..255 (8 wave32s)
    const int tj = blockIdx.x;           // output column tile 0..36
    const int ti = blockIdx.y;           // output row    tile 0..36

    // Contributing source blocks for this tile (uniform per workgroup)
    const int ilo = (ti - 5 > 0) ? ti - 5 : 0;
    const int ihi = (ti < NGRID - 1) ? ti : NGRID - 1;
    const int jlo = (tj - 5 > 0) ? tj - 5 : 0;
    const int jhi = (tj < NGRID - 1) ? tj : NGRID - 1;
    const int nJ  = jhi - jlo + 1;
    const int S   = (ihi - ilo + 1) * nJ;           // 1..36 stages

    const uint64_t eyBase  = (uint64_t)(uintptr_t)Ey;
    // Low 32 bits of a generic pointer to LDS == workgroup-relative LDS byte
    // offset, which is exactly what the async-load VDST operand wants.
    const uint32_t ldsBase = (uint32_t)(uintptr_t)(&smem[0]);

    // Per-thread 16B chunk decode (4 chunks/thread x 256 threads = 16 KB stage)
    uint32_t gOffLocal[4];   // byte offset within a stage's Ey region
    uint32_t ldsOff[4];      // byte offset within a stage buffer
    #pragma unroll
    for (int q = 0; q < 4; ++q) {
        const int idx = t + q * 256;       // 16B chunk id 0..1023
        const int e   = idx * 4;           // float id 0..4095
        const int m   = e >> 10;           // mode
        const int rr  = (e >> 5) & 31;     // row within 32x32 subtile
        const int yy  = e & 31;            // col within subtile (mult of 4)
        gOffLocal[q] = (uint32_t)(m * EY_PL + rr * EY_SZ + yy) * 4u;
        ldsOff[q]    = (uint32_t)idx * 16u;
    }

    // Issue one stage's 4 async B128 loads per thread (wave: +4 on ASYNCcnt).
    // NT hint: Ey is 604 MB streamed exactly once -> don't churn the 192MB L2.
    auto issue = [&](int s, int buf) {
        const int i = ilo + s / nJ;
        const int j = jlo + s % nJ;
        const uint32_t base =
            (uint32_t)((i * NGRID + j) * MODES * EY_PL      // k*4 planes
                       + (ti - i) * OUT_RES * EY_SZ         // x0 * 192
                       + (tj - j) * OUT_RES) * 4u;          // y0
        const uint32_t lbuf = ldsBase + (uint32_t)buf * STAGE_BYTES;
        #pragma unroll
        for (int q = 0; q < 4; ++q) {
            const uint32_t go = base + gOffLocal[q];
            const uint32_t la = lbuf + ldsOff[q];
            asm volatile("global_load_async_to_lds_b128 %0, %1, %2 th:TH_LOAD_NT"
                         :: "v"(la), "v"(go), "s"(eyBase)
                         : "memory");
        }
    };

    // Output pixel ownership: thread t -> (row ty, cols tx..tx+3)
    const int ty = t >> 3;            // 0..31
    const int tx = (t & 7) * 4;       // 0,4,...,28
    float4 acc = make_float4(0.f, 0.f, 0.f, 0.f);

    // Software pipeline: keep up to 2 stages (8 async B128 per wave) in flight.
    issue(0, 0);
    if (S > 1) issue(1, 1);
    for (int s = 0; s < S; ++s) {
        const int buf = s % NBUF;
        if (s + 2 < S) issue(s + 2, (s + 2) % NBUF);

        // Async completion is in issue order: with `ahead` stages still
        // outstanding (4 instrs each), waiting to <=4*ahead means stage s
        // has fully landed in LDS.
        const int ahead = (S - 1 - s > 2) ? 2 : (S - 1 - s);
        if (ahead == 2)      asm volatile("s_wait_asynccnt 8" ::: "memory");
        else if (ahead == 1) asm volatile("s_wait_asynccnt 4" ::: "memory");
        else                 asm volatile("s_wait_asynccnt 0" ::: "memory");
        __syncthreads();   // stage s visible to all 8 waves

        const int i  = ilo + s / nJ;
        const int j  = jlo + s % nJ;
        const int k4 = (i * NGRID + j) * MODES;
        const float* bufp = &smem[buf * STAGE_FLOATS];
        #pragma unroll
        for (int m = 0; m < MODES; ++m) {
            const float nf = neff[k4 + m];                       // uniform -> scalar
            const float w  = U[k4 + m] * (nf * 1.5f / (nf + 1.5f));
            const float4 v = *(const float4*)&bufp[(m * 32 + ty) * 32 + tx];
            acc.x += w * v.x;
            acc.y += w * v.y;
            acc.z += w * v.z;
            acc.w += w * v.w;
        }
        __syncthreads();   // all readers done before this buffer is reused (s+3)
    }

    const int r = ti * 32 + ty;
    const int c = tj * 32 + tx;
    *(float4*)&En[(size_t)r * TOTAL + c] = acc;
}

extern "C" void kernel_launch(void* const* d_in, const int* in_sizes, int n_in,
                              void* d_out, int out_size, void* d_ws, size_t ws_size,
                              hipStream_t stream) {
    (void)in_sizes; (void)n_in; (void)d_ws; (void)ws_size; (void)out_size;
    // setup_inputs order: hs (unused by reference), U, neff, Ey
    const float* U    = (const float*)d_in[1];
    const float* neff = (const float*)d_in[2];
    const float* Ey   = (const float*)d_in[3];
    float* En = (float*)d_out;

    dim3 grid(NTILE, NTILE);   // 37 x 37 output tiles
    dim3 block(256);           // 8 wave32s
    gen_en_tile_kernel<<<grid, block, 0, stream>>>(U, neff, Ey, En);
}